// EGConv_936302871065
// MI455X (gfx1250) — compile-verified
//
#include <hip/hip_runtime.h>

#define N_NODES 100000
#define N_EDGES 1250000
#define D 64            // IN_CH == OUT_CH
#define FEATS 192       // 3*D

typedef __attribute__((ext_vector_type(2))) float v2f;
typedef __attribute__((ext_vector_type(8))) float v8f;

#define NEG_FLT_MAX (-3.402823466e+38f)

// ---------------------------------------------------------------------------
// Kernel 0: init workspace (agg_sum = 0, agg_max = -FLT_MAX, count = 0)
// ---------------------------------------------------------------------------
__global__ void egconv_init(float* __restrict__ agg_sum,
                            float* __restrict__ agg_max,
                            float* __restrict__ cnt) {
    int i = blockIdx.x * blockDim.x + threadIdx.x;
    if (i < N_NODES * D) {
        agg_sum[i] = 0.0f;
        agg_max[i] = NEG_FLT_MAX;
    }
    if (i < N_NODES) cnt[i] = 0.0f;
}

// Monotonic-bit float max: native int atomics (no CAS loop).
__device__ inline void atomicMaxF32(float* addr, float v) {
    if (v >= 0.0f) {
        atomicMax((int*)addr, __float_as_int(v));
    } else {
        atomicMin((unsigned int*)addr, __float_as_uint(v));
    }
}

// ---------------------------------------------------------------------------
// Kernel 1: edge scatter. One wave32 per edge; each lane owns 2 channels
// (float2 -> one coalesced 256B load of x[src] per edge). sum/max atomics
// land in L2 (all aggregation buffers are L2-resident on MI455X: 51MB << 192MB).
// ---------------------------------------------------------------------------
__global__ void egconv_scatter(const float* __restrict__ x,
                               const int*  __restrict__ src,
                               const int*  __restrict__ dst,
                               float* __restrict__ agg_sum,
                               float* __restrict__ agg_max,
                               float* __restrict__ cnt) {
    const int lane = threadIdx.x & 31;
    const int eid  = blockIdx.x * (blockDim.x >> 5) + (threadIdx.x >> 5);
    if (eid >= N_EDGES) return;

    const int s = src[eid];
    const int d = dst[eid];

    const int c = lane * 2;
    const v2f xv = *(const v2f*)(x + (size_t)s * D + c);

    float* ps = agg_sum + (size_t)d * D + c;
    float* pm = agg_max + (size_t)d * D + c;
    atomicAdd(ps + 0, xv.x);
    atomicAdd(ps + 1, xv.y);
    atomicMaxF32(pm + 0, xv.x);
    atomicMaxF32(pm + 1, xv.y);
    if (lane == 0) atomicAdd(cnt + d, 1.0f);
}

// ---------------------------------------------------------------------------
// Kernel 2: fused feats construction + GEMM via V_WMMA_F32_16X16X4_F32.
// Block = 128 threads (4 waves). Block owns 16 nodes; builds the 16x192
// feats tile [sum | mean | max-masked] in LDS, then wave w computes the
// 16(node) x 16(out-ch) tile for o_base = 16*w with 48 chained f32 WMMAs.
//
// VGPR layouts per CDNA5 ISA 7.12.2:
//  A 16x4 f32 : lanes 0-15 row M=lane, VGPR0=K0,VGPR1=K1; lanes 16-31 K2,K3
//  B 4x16 f32 : lanes 0-15 col N=lane, VGPR0=K0,VGPR1=K1; lanes 16-31 K2,K3
//  C/D 16x16  : VGPR r -> M=r (lanes 0-15) / M=r+8 (lanes 16-31), N=lane&15
// ---------------------------------------------------------------------------
__global__ void egconv_gemm(const float* __restrict__ agg_sum,
                            const float* __restrict__ agg_max,
                            const float* __restrict__ cnt,
                            const float* __restrict__ W,   // [64, 192] row-major
                            const float* __restrict__ bias,
                            float* __restrict__ out) {     // [N, 64]
    __shared__ float feats[16 * FEATS];   // 12 KB

    const int node_base = blockIdx.x * 16;
    const int tid = threadIdx.x;

    // Stage 16x192 feats tile: each thread handles 8 (row, channel) pairs.
    for (int i = tid; i < 16 * D; i += 128) {
        const int row = i >> 6;          // 0..15
        const int c   = i & (D - 1);     // 0..63
        const int node = node_base + row;
        const float s  = agg_sum[(size_t)node * D + c];
        const float n  = cnt[node];
        const float mean = s / fmaxf(n, 1.0f);
        const float mx = (n > 0.0f) ? agg_max[(size_t)node * D + c] : 0.0f;
        float* fr = feats + row * FEATS;
        fr[c]           = s;
        fr[D + c]       = mean;
        fr[2 * D + c]   = mx;
    }
    __syncthreads();

    const int wave   = tid >> 5;
    const int lane   = tid & 31;
    const int l15    = lane & 15;
    const int hi     = lane >> 4;        // half-wave select
    const int klo    = hi * 2;           // K offset 0 or 2 within the K=4 step
    const int o_base = wave * 16;

    const float* wrow = W + (size_t)(o_base + l15) * FEATS;  // B column = out-ch
    const float* arow = feats + l15 * FEATS;                 // A row   = node

    v8f c8 = {};
#pragma unroll
    for (int k0 = 0; k0 < FEATS; k0 += 4) {
        v2f a, b;
        a.x = arow[k0 + klo];
        a.y = arow[k0 + klo + 1];
        b.x = wrow[k0 + klo];
        b.y = wrow[k0 + klo + 1];
        c8 = __builtin_amdgcn_wmma_f32_16x16x4_f32(
                 /*neg_a=*/false, a, /*neg_b=*/false, b,
                 /*c_mod=*/(short)0, c8, /*reuse_a=*/false, /*reuse_b=*/false);
    }

    const float bv = bias[o_base + l15];
#pragma unroll
    for (int r = 0; r < 8; ++r) {
        const int node = node_base + r + hi * 8;
        out[(size_t)node * D + o_base + l15] = c8[r] + bv;
    }
}

// ---------------------------------------------------------------------------
extern "C" void kernel_launch(void* const* d_in, const int* in_sizes, int n_in,
                              void* d_out, int out_size, void* d_ws, size_t ws_size,
                              hipStream_t stream) {
    const float* x    = (const float*)d_in[0];
    const int*   ei   = (const int*)d_in[1];      // [2, E]
    const float* W    = (const float*)d_in[2];    // [64, 192]
    const float* bias = (const float*)d_in[3];    // [64]
    float*       out  = (float*)d_out;

    const int* src = ei;
    const int* dst = ei + N_EDGES;

    float* agg_sum = (float*)d_ws;                       // N*64
    float* agg_max = agg_sum + (size_t)N_NODES * D;      // N*64
    float* cnt     = agg_max + (size_t)N_NODES * D;      // N

    // Phase 0: init accumulators.
    {
        const int total = N_NODES * D;
        egconv_init<<<(total + 255) / 256, 256, 0, stream>>>(agg_sum, agg_max, cnt);
    }
    // Phase 1: edge scatter (wave per edge, 8 edges per 256-thread block).
    {
        const int edges_per_block = 256 / 32;
        const int blocks = (N_EDGES + edges_per_block - 1) / edges_per_block;
        egconv_scatter<<<blocks, 256, 0, stream>>>(x, src, dst, agg_sum, agg_max, cnt);
    }
    // Phase 2: fused feats + WMMA GEMM (16 nodes per block, 4 waves x 16 out-ch).
    {
        const int blocks = N_NODES / 16;  // 6250
        egconv_gemm<<<blocks, 128, 0, stream>>>(agg_sum, agg_max, cnt, W, bias, out);
    }
}